// PointSetDifferenceModule_22162031247560
// MI455X (gfx1250) — compile-verified
//
#include <hip/hip_runtime.h>

// ============================================================================
// PointSetDifference for MI455X (gfx1250, wave32, WMMA f32 16x16x4)
//
// BN folded analytically: stats of h = x@W1^T+b1 derived from moments
//   S = sum(x x^T), s = sum(x)  ->  mu, var per channel -> fold into W1.
// MLP GEMMs and diff-sample S = X^T X run on v_wmma_f32_16x16x4_f32; agg/comb S on VALU.
// ============================================================================

#define B_  4
#define N_  8192
#define C_  64
#define K_  16
#define P_  (B_ * N_)                 // 32768 points per set
#define EPS_ 1e-5f

typedef __attribute__((ext_vector_type(2))) float v2f;
typedef __attribute__((ext_vector_type(8))) float v8f;

__device__ __forceinline__ v8f wmma4(v2f a, v2f b, v8f c) {
  // D(16x16,f32) = A(16x4,f32) x B(4x16,f32) + C
  return __builtin_amdgcn_wmma_f32_16x16x4_f32(false, a, false, b, (short)0, c,
                                               false, false);
}
__device__ __forceinline__ v8f zero8() {
  v8f z = {0.f, 0.f, 0.f, 0.f, 0.f, 0.f, 0.f, 0.f};
  return z;
}

// ----------------------------------------------------------------------------
// Moments of diff samples: S(64x64) += x x^T, sum(64) += x, over all (b,n,k),
// x = fc[b,n] - fn[b, idx[b,n,k]].  X staged transposed in LDS: X[c*18 + m].
// A-frag (channels x samples) and B-frag (samples x channels) of the same X
// chunk have identical per-lane register contents -> one fragment, two roles.
// ----------------------------------------------------------------------------
__global__ void __launch_bounds__(256, 1)
diff_moments_kernel(const float* __restrict__ fc, const float* __restrict__ fn,
                    const int* __restrict__ idx,
                    float* __restrict__ Sg, float* __restrict__ sumg, int P) {
  __shared__ float Xl[8][64 * 18];
  __shared__ float Sacc[64 * 64];
  __shared__ float sumAcc[64];
  for (int i = threadIdx.x; i < 64 * 64; i += blockDim.x) Sacc[i] = 0.f;
  if (threadIdx.x < 64) sumAcc[threadIdx.x] = 0.f;
  __syncthreads();

  const int lane = threadIdx.x & 31, wid = threadIdx.x >> 5;
  const int l15 = lane & 15, half = lane >> 4;
  float* X = Xl[wid];

  v8f sacc[4][4];
#pragma unroll
  for (int r = 0; r < 4; ++r)
#pragma unroll
    for (int q = 0; q < 4; ++q) sacc[r][q] = zero8();
  float sx0 = 0.f, sx1 = 0.f;

  const int gw = blockIdx.x * 8 + wid, nw = gridDim.x * 8;
  for (int p = gw; p < P; p += nw) {
    const int bbase = (p >> 13) << 13;  // b * N
    v2f cen = *(const v2f*)(fc + (size_t)p * 64 + 2 * lane);
#pragma unroll
    for (int m = 0; m < 16; ++m) {
      int nb = idx[(size_t)p * 16 + m];
      v2f nv = *(const v2f*)(fn + (size_t)(bbase + nb) * 64 + 2 * lane);
      float d0 = cen.x - nv.x, d1 = cen.y - nv.y;
      X[(2 * lane) * 18 + m] = d0;
      X[(2 * lane + 1) * 18 + m] = d1;
      sx0 += d0; sx1 += d1;
    }
    // wave-private LDS region: per-wave DS ordering suffices, no barrier
#pragma unroll
    for (int t = 0; t < 4; ++t) {
      v2f f[4];
#pragma unroll
      for (int r = 0; r < 4; ++r)
        f[r] = *(const v2f*)&X[(r * 16 + l15) * 18 + t * 4 + half * 2];
#pragma unroll
      for (int r = 0; r < 4; ++r)
#pragma unroll
        for (int q = 0; q < 4; ++q) sacc[r][q] = wmma4(f[r], f[q], sacc[r][q]);
    }
  }
#pragma unroll
  for (int r = 0; r < 4; ++r)
#pragma unroll
    for (int q = 0; q < 4; ++q)
#pragma unroll
      for (int e = 0; e < 8; ++e) {
        int row = r * 16 + e + half * 8, col = q * 16 + l15;
        atomicAdd(&Sacc[row * 64 + col], sacc[r][q][e]);
      }
  atomicAdd(&sumAcc[2 * lane], sx0);
  atomicAdd(&sumAcc[2 * lane + 1], sx1);
  __syncthreads();
  for (int i = threadIdx.x; i < 64 * 64; i += blockDim.x) atomicAdd(&Sg[i], Sacc[i]);
  if (threadIdx.x < 64) atomicAdd(&sumg[threadIdx.x], sumAcc[threadIdx.x]);
}

// ----------------------------------------------------------------------------
// Moments of a materialized matrix X (P x CIN): per-thread TRxTR tile of S.
// ----------------------------------------------------------------------------
template <int CIN>
__global__ void __launch_bounds__(256, 1)
mat_moments_kernel(const float* __restrict__ X, float* __restrict__ Sg,
                   float* __restrict__ sumg, int P) {
  const int TR = CIN / 16;
  __shared__ float xs[8][CIN];
  const int t = threadIdx.x;
  const int r0 = (t >> 4) * TR, c0 = (t & 15) * TR;
  float acc[TR][TR];
#pragma unroll
  for (int i = 0; i < TR; ++i)
#pragma unroll
    for (int j = 0; j < TR; ++j) acc[i][j] = 0.f;
  float sx = 0.f;

  for (int s0 = blockIdx.x * 8; s0 < P; s0 += gridDim.x * 8) {
    for (int i = t; i < 8 * CIN; i += 256) {
      int sm = i / CIN, c = i % CIN;
      xs[sm][c] = X[(size_t)(s0 + sm) * CIN + c];
    }
    __syncthreads();
#pragma unroll
    for (int sm = 0; sm < 8; ++sm) {
      float xr[TR], xc[TR];
#pragma unroll
      for (int i = 0; i < TR; ++i) { xr[i] = xs[sm][r0 + i]; xc[i] = xs[sm][c0 + i]; }
#pragma unroll
      for (int i = 0; i < TR; ++i)
#pragma unroll
        for (int j = 0; j < TR; ++j) acc[i][j] += xr[i] * xc[j];
      if (t < CIN) sx += xs[sm][t];
    }
    __syncthreads();
  }
#pragma unroll
  for (int i = 0; i < TR; ++i)
#pragma unroll
    for (int j = 0; j < TR; ++j)
      atomicAdd(&Sg[(size_t)(r0 + i) * CIN + c0 + j], acc[i][j]);
  if (t < CIN) atomicAdd(&sumg[t], sx);
}

// ----------------------------------------------------------------------------
// Fold BN(h = x@w1^T + b1) into the first linear using moments.
// ----------------------------------------------------------------------------
__global__ void finalize_kernel(const float* __restrict__ S, const float* __restrict__ sum,
                                const float* __restrict__ w1, const float* __restrict__ b1,
                                const float* __restrict__ g, const float* __restrict__ beta,
                                float* __restrict__ W1s, float* __restrict__ b1s,
                                int Cin, int Cout, float invCnt) {
  int j = threadIdx.x;
  if (j >= Cout) return;
  const float* wj = w1 + (size_t)j * Cin;
  float mu0 = 0.f;
  for (int i = 0; i < Cin; ++i) mu0 += wj[i] * sum[i];
  mu0 *= invCnt;
  float mu = mu0 + b1[j];
  float q = 0.f;
  for (int i1 = 0; i1 < Cin; ++i1) {
    const float* Srow = S + (size_t)i1 * Cin;
    float a = 0.f;
    for (int i2 = 0; i2 < Cin; ++i2) a += Srow[i2] * wj[i2];
    q += wj[i1] * a;
  }
  q *= invCnt;
  float Ehh = q + 2.f * b1[j] * mu0 + b1[j] * b1[j];
  float var = Ehh - mu * mu;
  float sc = g[j] * rsqrtf(var + EPS_);
  for (int i = 0; i < Cin; ++i) W1s[(size_t)j * Cin + i] = wj[i] * sc;
  b1s[j] = b1[j] * sc + (beta[j] - mu * sc);
}

// ----------------------------------------------------------------------------
// Diff MLP (fused): per point, X = center - neighbors (16x64) ->
// relu(X@W1s^T + b1s) @ W2^T + b2 -> max over 16 rows -> comb[:, 0:64].
// Weights register-resident as B-fragments; X row-major LDS, stride 68.
// ----------------------------------------------------------------------------
__global__ void __launch_bounds__(256, 1)
diff_mlp_kernel(const float* __restrict__ fc, const float* __restrict__ fn,
                const int* __restrict__ idx,
                const float* __restrict__ W1s, const float* __restrict__ b1s,
                const float* __restrict__ W2, const float* __restrict__ b2,
                float* __restrict__ comb, int P) {
  __shared__ float Xl[8][16 * 68];
  const int lane = threadIdx.x & 31, wid = threadIdx.x >> 5;
  const int l15 = lane & 15, half = lane >> 4;
  float* X = Xl[wid];

  v2f w1f[4][16], w2f[4][16];
  float b1v[4], b2v[4];
#pragma unroll
  for (int j = 0; j < 4; ++j) {
    int n = j * 16 + l15;
    b1v[j] = b1s[n]; b2v[j] = b2[n];
#pragma unroll
    for (int s = 0; s < 16; ++s) {
      int k = s * 4 + half * 2;
      w1f[j][s] = *(const v2f*)(W1s + (size_t)n * 64 + k);
      w2f[j][s] = *(const v2f*)(W2 + (size_t)n * 64 + k);
    }
  }

  const int gw = blockIdx.x * 8 + wid, nw = gridDim.x * 8;
  for (int p = gw; p < P; p += nw) {
    const int bbase = (p >> 13) << 13;
    v2f cen = *(const v2f*)(fc + (size_t)p * 64 + 2 * lane);
#pragma unroll
    for (int m = 0; m < 16; ++m) {
      int nb = idx[(size_t)p * 16 + m];
      v2f nv = *(const v2f*)(fn + (size_t)(bbase + nb) * 64 + 2 * lane);
      v2f d; d.x = cen.x - nv.x; d.y = cen.y - nv.y;
      *(v2f*)&X[m * 68 + 2 * lane] = d;
    }
    v2f af[16];
#pragma unroll
    for (int s = 0; s < 16; ++s)
      af[s] = *(const v2f*)&X[l15 * 68 + s * 4 + half * 2];

    v8f acc[4];
#pragma unroll
    for (int j = 0; j < 4; ++j) acc[j] = zero8();
#pragma unroll
    for (int j = 0; j < 4; ++j)
#pragma unroll
      for (int s = 0; s < 16; ++s) acc[j] = wmma4(af[s], w1f[j][s], acc[j]);

#pragma unroll
    for (int j = 0; j < 4; ++j)
#pragma unroll
      for (int e = 0; e < 8; ++e)
        X[(e + half * 8) * 68 + j * 16 + l15] = fmaxf(acc[j][e] + b1v[j], 0.f);
#pragma unroll
    for (int s = 0; s < 16; ++s)
      af[s] = *(const v2f*)&X[l15 * 68 + s * 4 + half * 2];

    v8f acc2[4];
#pragma unroll
    for (int j = 0; j < 4; ++j) acc2[j] = zero8();
#pragma unroll
    for (int j = 0; j < 4; ++j)
#pragma unroll
      for (int s = 0; s < 16; ++s) acc2[j] = wmma4(af[s], w2f[j][s], acc2[j]);

#pragma unroll
    for (int j = 0; j < 4; ++j) {
      float mx = acc2[j][0];
#pragma unroll
      for (int e = 1; e < 8; ++e) mx = fmaxf(mx, acc2[j][e]);
      mx += b2v[j];
      float o = fmaxf(mx, __shfl_xor(mx, 16, 32));
      if (half == 0) comb[(size_t)p * 128 + j * 16 + l15] = o;
    }
  }
}

// ----------------------------------------------------------------------------
// Softmax-weighted neighbor aggregation (VALU): agg[p] = softmax(f.nbr) @ nbr.
// ----------------------------------------------------------------------------
__global__ void __launch_bounds__(256, 1)
sim_agg_kernel(const float* __restrict__ fc, const float* __restrict__ fn,
               const int* __restrict__ idx, float* __restrict__ agg, int P) {
  const int lane = threadIdx.x & 31, wid = threadIdx.x >> 5;
  const int gw = blockIdx.x * 8 + wid, nw = gridDim.x * 8;
  for (int p = gw; p < P; p += nw) {
    const int bbase = (p >> 13) << 13;
    v2f cen = *(const v2f*)(fc + (size_t)p * 64 + 2 * lane);
    float dots[16];
    v2f nvs[16];
#pragma unroll
    for (int k = 0; k < 16; ++k) {
      int nb = idx[(size_t)p * 16 + k];
      nvs[k] = *(const v2f*)(fn + (size_t)(bbase + nb) * 64 + 2 * lane);
      float part = cen.x * nvs[k].x + cen.y * nvs[k].y;
#pragma unroll
      for (int off = 16; off; off >>= 1) part += __shfl_xor(part, off, 32);
      dots[k] = part;
    }
    float mx = dots[0];
#pragma unroll
    for (int k = 1; k < 16; ++k) mx = fmaxf(mx, dots[k]);
    float es[16], se = 0.f;
#pragma unroll
    for (int k = 0; k < 16; ++k) { es[k] = __expf(dots[k] - mx); se += es[k]; }
    float inv = 1.f / se;
    float ax = 0.f, ay = 0.f;
#pragma unroll
    for (int k = 0; k < 16; ++k) {
      float w = es[k] * inv;
      ax += w * nvs[k].x; ay += w * nvs[k].y;
    }
    v2f o; o.x = ax; o.y = ay;
    *(v2f*)(agg + (size_t)p * 64 + 2 * lane) = o;
  }
}

// ----------------------------------------------------------------------------
// Sim MLP (fused, 16 points per wave): comb[:, 64:128].
// ----------------------------------------------------------------------------
__global__ void __launch_bounds__(256, 1)
sim_mlp_kernel(const float* __restrict__ agg,
               const float* __restrict__ W1s, const float* __restrict__ b1s,
               const float* __restrict__ W2, const float* __restrict__ b2,
               float* __restrict__ comb, int P) {
  __shared__ float Xl[8][16 * 68];
  const int lane = threadIdx.x & 31, wid = threadIdx.x >> 5;
  const int l15 = lane & 15, half = lane >> 4;
  float* X = Xl[wid];

  v2f w1f[4][16], w2f[4][16];
  float b1v[4], b2v[4];
#pragma unroll
  for (int j = 0; j < 4; ++j) {
    int n = j * 16 + l15;
    b1v[j] = b1s[n]; b2v[j] = b2[n];
#pragma unroll
    for (int s = 0; s < 16; ++s) {
      int k = s * 4 + half * 2;
      w1f[j][s] = *(const v2f*)(W1s + (size_t)n * 64 + k);
      w2f[j][s] = *(const v2f*)(W2 + (size_t)n * 64 + k);
    }
  }

  const int gw = blockIdx.x * 8 + wid, nw = gridDim.x * 8;
  const int T = P >> 4;
  for (int tile = gw; tile < T; tile += nw) {
    const int p0 = tile << 4;
#pragma unroll
    for (int m = 0; m < 16; ++m) {
      v2f a = *(const v2f*)(agg + (size_t)(p0 + m) * 64 + 2 * lane);
      *(v2f*)&X[m * 68 + 2 * lane] = a;
    }
    v2f af[16];
#pragma unroll
    for (int s = 0; s < 16; ++s)
      af[s] = *(const v2f*)&X[l15 * 68 + s * 4 + half * 2];
    v8f acc[4];
#pragma unroll
    for (int j = 0; j < 4; ++j) acc[j] = zero8();
#pragma unroll
    for (int j = 0; j < 4; ++j)
#pragma unroll
      for (int s = 0; s < 16; ++s) acc[j] = wmma4(af[s], w1f[j][s], acc[j]);
#pragma unroll
    for (int j = 0; j < 4; ++j)
#pragma unroll
      for (int e = 0; e < 8; ++e)
        X[(e + half * 8) * 68 + j * 16 + l15] = fmaxf(acc[j][e] + b1v[j], 0.f);
#pragma unroll
    for (int s = 0; s < 16; ++s)
      af[s] = *(const v2f*)&X[l15 * 68 + s * 4 + half * 2];
    v8f acc2[4];
#pragma unroll
    for (int j = 0; j < 4; ++j) acc2[j] = zero8();
#pragma unroll
    for (int j = 0; j < 4; ++j)
#pragma unroll
      for (int s = 0; s < 16; ++s) acc2[j] = wmma4(af[s], w2f[j][s], acc2[j]);
#pragma unroll
    for (int j = 0; j < 4; ++j)
#pragma unroll
      for (int e = 0; e < 8; ++e)
        comb[(size_t)(p0 + e + half * 8) * 128 + 64 + j * 16 + l15] =
            acc2[j][e] + b2v[j];
  }
}

// ----------------------------------------------------------------------------
// Final MLP (fused, 16 points per wave): 128 -> 128 (BN-folded, relu) -> 64.
// Weight B-fragments streamed from global (L1/L2 resident); X in LDS, stride 130.
// ----------------------------------------------------------------------------
__global__ void __launch_bounds__(128, 1)
fin_mlp_kernel(const float* __restrict__ comb,
               const float* __restrict__ W1s, const float* __restrict__ b1s,
               const float* __restrict__ W2, const float* __restrict__ b2,
               float* __restrict__ out, int P) {
  __shared__ float Xl[4][16 * 130];
  const int lane = threadIdx.x & 31, wid = threadIdx.x >> 5;
  const int l15 = lane & 15, half = lane >> 4;
  float* X = Xl[wid];

  float b1v[8], b2v[4];
#pragma unroll
  for (int j = 0; j < 8; ++j) b1v[j] = b1s[j * 16 + l15];
#pragma unroll
  for (int j = 0; j < 4; ++j) b2v[j] = b2[j * 16 + l15];

  const int gw = blockIdx.x * 4 + wid, nw = gridDim.x * 4;
  const int T = P >> 4;
  for (int tile = gw; tile < T; tile += nw) {
    const int p0 = tile << 4;
#pragma unroll
    for (int m = 0; m < 16; ++m) {
      v2f a = *(const v2f*)(comb + (size_t)(p0 + m) * 128 + 2 * lane);
      v2f b = *(const v2f*)(comb + (size_t)(p0 + m) * 128 + 64 + 2 * lane);
      *(v2f*)&X[m * 130 + 2 * lane] = a;
      *(v2f*)&X[m * 130 + 64 + 2 * lane] = b;
    }
    v2f af[32];
#pragma unroll
    for (int s = 0; s < 32; ++s)
      af[s] = *(const v2f*)&X[l15 * 130 + s * 4 + half * 2];

    v8f acc[8];
#pragma unroll
    for (int j = 0; j < 8; ++j) acc[j] = zero8();
#pragma unroll
    for (int j = 0; j < 8; ++j) {
      const float* wrow = W1s + (size_t)(j * 16 + l15) * 128 + half * 2;
#pragma unroll
      for (int s = 0; s < 32; ++s) {
        v2f bf = *(const v2f*)(wrow + s * 4);
        acc[j] = wmma4(af[s], bf, acc[j]);
      }
    }
#pragma unroll
    for (int j = 0; j < 8; ++j)
#pragma unroll
      for (int e = 0; e < 8; ++e)
        X[(e + half * 8) * 130 + j * 16 + l15] = fmaxf(acc[j][e] + b1v[j], 0.f);
#pragma unroll
    for (int s = 0; s < 32; ++s)
      af[s] = *(const v2f*)&X[l15 * 130 + s * 4 + half * 2];

    v8f acc2[4];
#pragma unroll
    for (int j = 0; j < 4; ++j) acc2[j] = zero8();
#pragma unroll
    for (int j = 0; j < 4; ++j) {
      const float* wrow = W2 + (size_t)(j * 16 + l15) * 128 + half * 2;
#pragma unroll
      for (int s = 0; s < 32; ++s) {
        v2f bf = *(const v2f*)(wrow + s * 4);
        acc2[j] = wmma4(af[s], bf, acc2[j]);
      }
    }
#pragma unroll
    for (int j = 0; j < 4; ++j)
#pragma unroll
      for (int e = 0; e < 8; ++e)
        out[(size_t)(p0 + e + half * 8) * 64 + j * 16 + l15] = acc2[j][e] + b2v[j];
  }
}

// ============================================================================
extern "C" void kernel_launch(void* const* d_in, const int* in_sizes, int n_in,
                              void* d_out, int out_size, void* d_ws, size_t ws_size,
                              hipStream_t stream) {
  (void)in_sizes; (void)n_in; (void)out_size; (void)ws_size;
  const float* f0  = (const float*)d_in[0];
  const float* f1  = (const float*)d_in[1];
  const int*   i01 = (const int*)d_in[2];
  const int*   i10 = (const int*)d_in[3];
  const float* dw1 = (const float*)d_in[4];
  const float* db1 = (const float*)d_in[5];
  const float* dg  = (const float*)d_in[6];
  const float* dbe = (const float*)d_in[7];
  const float* dw2 = (const float*)d_in[8];
  const float* db2 = (const float*)d_in[9];
  const float* sw1 = (const float*)d_in[10];
  const float* sb1 = (const float*)d_in[11];
  const float* sg  = (const float*)d_in[12];
  const float* sbe = (const float*)d_in[13];
  const float* sw2 = (const float*)d_in[14];
  const float* sb2 = (const float*)d_in[15];
  const float* fw1 = (const float*)d_in[16];
  const float* fb1 = (const float*)d_in[17];
  const float* fg  = (const float*)d_in[18];
  const float* fbe = (const float*)d_in[19];
  const float* fw2 = (const float*)d_in[20];
  const float* fb2 = (const float*)d_in[21];

  float* w = (float*)d_ws;
  size_t o = 0;
  auto alloc = [&](size_t n) { float* p = w + o; o += n; return p; };
  float* Sd[2] = {alloc(4096), alloc(4096)};
  float* sd[2] = {alloc(64), alloc(64)};
  float* Ss[2] = {alloc(4096), alloc(4096)};
  float* ss[2] = {alloc(64), alloc(64)};
  float* Sf[2] = {alloc(16384), alloc(16384)};
  float* sf[2] = {alloc(128), alloc(128)};
  const size_t statsFloats = o;
  float* dW1s[2] = {alloc(4096), alloc(4096)};
  float* db1f[2] = {alloc(64), alloc(64)};
  float* sW1s[2] = {alloc(4096), alloc(4096)};
  float* sb1f[2] = {alloc(64), alloc(64)};
  float* fW1s[2] = {alloc(16384), alloc(16384)};
  float* fb1f[2] = {alloc(128), alloc(128)};
  float* agg[2]  = {alloc((size_t)P_ * 64), alloc((size_t)P_ * 64)};
  float* comb[2] = {alloc((size_t)P_ * 128), alloc((size_t)P_ * 128)};

  hipMemsetAsync(w, 0, statsFloats * sizeof(float), stream);

  const float* fc[2] = {f0, f1};
  const float* fn[2] = {f1, f0};
  const int* ix[2] = {i01, i10};
  float* outp[2] = {(float*)d_out, (float*)d_out + (size_t)P_ * 64};

  const float invDiff = 1.f / (float)(P_ * K_);
  const float invPts = 1.f / (float)P_;

  for (int d = 0; d < 2; ++d)
    diff_moments_kernel<<<128, 256, 0, stream>>>(fc[d], fn[d], ix[d], Sd[d], sd[d], P_);
  for (int d = 0; d < 2; ++d)
    finalize_kernel<<<1, 64, 0, stream>>>(Sd[d], sd[d], dw1, db1, dg, dbe,
                                          dW1s[d], db1f[d], 64, 64, invDiff);
  for (int d = 0; d < 2; ++d)
    diff_mlp_kernel<<<512, 256, 0, stream>>>(fc[d], fn[d], ix[d], dW1s[d], db1f[d],
                                             dw2, db2, comb[d], P_);
  for (int d = 0; d < 2; ++d)
    sim_agg_kernel<<<128, 256, 0, stream>>>(fc[d], fn[d], ix[d], agg[d], P_);
  for (int d = 0; d < 2; ++d)
    mat_moments_kernel<64><<<64, 256, 0, stream>>>(agg[d], Ss[d], ss[d], P_);
  for (int d = 0; d < 2; ++d)
    finalize_kernel<<<1, 64, 0, stream>>>(Ss[d], ss[d], sw1, sb1, sg, sbe,
                                          sW1s[d], sb1f[d], 64, 64, invPts);
  for (int d = 0; d < 2; ++d)
    sim_mlp_kernel<<<256, 256, 0, stream>>>(agg[d], sW1s[d], sb1f[d], sw2, sb2,
                                            comb[d], P_);
  for (int d = 0; d < 2; ++d)
    mat_moments_kernel<128><<<64, 256, 0, stream>>>(comb[d], Sf[d], sf[d], P_);
  for (int d = 0; d < 2; ++d)
    finalize_kernel<<<1, 128, 0, stream>>>(Sf[d], sf[d], fw1, fb1, fg, fbe,
                                           fW1s[d], fb1f[d], 128, 128, invPts);
  for (int d = 0; d < 2; ++d)
    fin_mlp_kernel<<<512, 128, 0, stream>>>(comb[d], fW1s[d], fb1f[d], fw2, fb2,
                                            outp[d], P_);
}